// HeteroGATEncoder_8083128451629
// MI455X (gfx1250) — compile-verified
//
#include <hip/hip_runtime.h>
#include <hip/hip_bf16.h>
#include <math.h>

#define N_NODES   100000
#define N_EDGES   1000000
#define R_REL     3
#define IN_DIM    128
#define OUT_DIM   64
#define NEG_SLOPE 0.2f

typedef __attribute__((ext_vector_type(2))) float v2f;
typedef __attribute__((ext_vector_type(8))) float v8f;

static __device__ __forceinline__ void atomicMaxF32(float* addr, float val) {
    // float atomic max via monotonic int/uint mapping (addr initialized to -inf)
    if (val >= 0.0f) atomicMax((int*)addr, __float_as_int(val));
    else             atomicMin((unsigned int*)addr, __float_as_uint(val));
}

// ---------------------------------------------------------------- init
__global__ void fill_f32(float* __restrict__ p, float v, int n) {
    int i = blockIdx.x * blockDim.x + threadIdx.x;
    if (i < n) p[i] = v;
}

// ---------------------------------------------------------------- GEMM (WMMA f32 16x16x4)
// H[N,64] = X[N,K] @ W[K,64].  One wave32 per 16-row tile; 4 column tiles of 16.
// A frag (16x4 f32):  lanes 0-15 -> M=lane, K={k,k+1}; lanes 16-31 -> M=lane-16, K={k+2,k+3}
// B frag (4x16 f32):  lanes 0-15 -> N=lane, K={k,k+1}; lanes 16-31 -> N=lane-16, K={k+2,k+3}
// C/D (16x16 f32):    vgpr v, lanes 0-15 -> (M=v, N=lane); lanes 16-31 -> (M=v+8, N=lane-16)
template <int K>
__global__ __launch_bounds__(256) void gemm_wmma_f32(const float* __restrict__ X,
                                                     const float* __restrict__ W,
                                                     float* __restrict__ H) {
    int wave = (blockIdx.x * blockDim.x + threadIdx.x) >> 5;
    int lane = threadIdx.x & 31;
    if (wave >= N_NODES / 16) return;

    const int row_base = wave * 16;
    const int hf = lane >> 4;      // which half-wave (K offset of 2)
    const int lr = lane & 15;

    v8f c0 = {}, c1 = {}, c2 = {}, c3 = {};
    const float* xrow = X + (size_t)(row_base + lr) * K;

#pragma unroll 4
    for (int k = 0; k < K; k += 4) {
        v2f a;
        a.x = xrow[k + 2 * hf];
        a.y = xrow[k + 1 + 2 * hf];
        const float* w0 = W + (size_t)(k + 2 * hf) * OUT_DIM;
        const float* w1 = w0 + OUT_DIM;

        v2f b;
        b.x = w0[lr];       b.y = w1[lr];
        c0 = __builtin_amdgcn_wmma_f32_16x16x4_f32(false, a, false, b, (short)0, c0, false, false);
        b.x = w0[16 + lr];  b.y = w1[16 + lr];
        c1 = __builtin_amdgcn_wmma_f32_16x16x4_f32(false, a, false, b, (short)0, c1, false, false);
        b.x = w0[32 + lr];  b.y = w1[32 + lr];
        c2 = __builtin_amdgcn_wmma_f32_16x16x4_f32(false, a, false, b, (short)0, c2, false, false);
        b.x = w0[48 + lr];  b.y = w1[48 + lr];
        c3 = __builtin_amdgcn_wmma_f32_16x16x4_f32(false, a, false, b, (short)0, c3, false, false);
    }

#pragma unroll
    for (int v = 0; v < 8; ++v) {
        float* dst = H + (size_t)(row_base + v + 8 * hf) * OUT_DIM;
        dst[lr]      = c0[v];
        dst[16 + lr] = c1[v];
        dst[32 + lr] = c2[v];
        dst[48 + lr] = c3[v];
    }
}

// ---------------------------------------------------------------- attention logits per node
// al_src[n] = h[n,:] . a_src ;  al_dst[n] = h[n,:] . a_dst   (one wave32 per node)
__global__ __launch_bounds__(256) void node_attn_logits(const float* __restrict__ H,
                                                        const float* __restrict__ a_src,
                                                        const float* __restrict__ a_dst,
                                                        float* __restrict__ al_src,
                                                        float* __restrict__ al_dst) {
    int node = (blockIdx.x * blockDim.x + threadIdx.x) >> 5;
    int lane = threadIdx.x & 31;
    if (node >= N_NODES) return;
    const float* h = H + (size_t)node * OUT_DIM;
    float h0 = h[lane], h1 = h[lane + 32];
    float s = h0 * a_src[lane] + h1 * a_src[lane + 32];
    float d = h0 * a_dst[lane] + h1 * a_dst[lane + 32];
#pragma unroll
    for (int m = 16; m >= 1; m >>= 1) {
        s += __shfl_xor(s, m, 32);
        d += __shfl_xor(d, m, 32);
    }
    if (lane == 0) { al_src[node] = s; al_dst[node] = d; }
}

// ---------------------------------------------------------------- edge pass 1: leaky-relu + segment max
__global__ __launch_bounds__(256) void edge_logits_max(const int* __restrict__ src,
                                                       const int* __restrict__ dst,
                                                       const float* __restrict__ al_src,
                                                       const float* __restrict__ al_dst,
                                                       float* __restrict__ e_buf,
                                                       float* __restrict__ mbuf) {
    int e = blockIdx.x * blockDim.x + threadIdx.x;
    if (e >= N_EDGES) return;
    int s = src[e], d = dst[e];
    float v = al_src[s] + al_dst[d];
    v = (v > 0.0f) ? v : NEG_SLOPE * v;
    e_buf[e] = v;
    atomicMaxF32(mbuf + d, v);
}

// ---------------------------------------------------------------- edge pass 2: exp + segment sum
__global__ __launch_bounds__(256) void edge_exp_sum(const int* __restrict__ dst,
                                                    const float* __restrict__ e_buf,
                                                    const float* __restrict__ mbuf,
                                                    float* __restrict__ num_buf,
                                                    float* __restrict__ den) {
    int e = blockIdx.x * blockDim.x + threadIdx.x;
    if (e >= N_EDGES) return;
    int d = dst[e];
    float num = __expf(e_buf[e] - mbuf[d]);
    num_buf[e] = num;
    atomicAdd(den + d, num);
}

// ---------------------------------------------------------------- edge pass 3: weighted scatter
// one wave32 per edge, lanes cover the 64 output features (2 each)
__global__ __launch_bounds__(256) void edge_scatter(const int* __restrict__ src,
                                                    const int* __restrict__ dst,
                                                    const float* __restrict__ num_buf,
                                                    const float* __restrict__ den,
                                                    const float* __restrict__ H,
                                                    float* __restrict__ acc) {
    int e = (blockIdx.x * blockDim.x + threadIdx.x) >> 5;
    int lane = threadIdx.x & 31;
    if (e >= N_EDGES) return;
    int s = src[e], d = dst[e];
    float coef = num_buf[e] / den[d];
    const float* h = H + (size_t)s * OUT_DIM;
    float* a = acc + (size_t)d * OUT_DIM;
    atomicAdd(a + lane,      coef * h[lane]);
    atomicAdd(a + lane + 32, coef * h[lane + 32]);
}

// ---------------------------------------------------------------- layer-1 epilogue: mean-over-R + l2norm + relu
__global__ __launch_bounds__(256) void post_norm_relu(const float* __restrict__ acc,
                                                      const float* __restrict__ b, // [R,OUT]
                                                      float* __restrict__ h1) {
    int node = (blockIdx.x * blockDim.x + threadIdx.x) >> 5;
    int lane = threadIdx.x & 31;
    if (node >= N_NODES) return;
    int c0 = lane, c1 = lane + 32;
    float bs0 = b[c0] + b[OUT_DIM + c0] + b[2 * OUT_DIM + c0];
    float bs1 = b[c1] + b[OUT_DIM + c1] + b[2 * OUT_DIM + c1];
    const float* a = acc + (size_t)node * OUT_DIM;
    float v0 = (a[c0] + bs0) * (1.0f / 3.0f);
    float v1 = (a[c1] + bs1) * (1.0f / 3.0f);
    float ss = v0 * v0 + v1 * v1;
#pragma unroll
    for (int m = 16; m >= 1; m >>= 1) ss += __shfl_xor(ss, m, 32);
    float inv = 1.0f / fmaxf(sqrtf(ss), 1e-12f);
    float* o = h1 + (size_t)node * OUT_DIM;
    o[c0] = fmaxf(v0 * inv, 0.0f);
    o[c1] = fmaxf(v1 * inv, 0.0f);
}

// ---------------------------------------------------------------- layer-2 epilogue: mean-over-R + bias
__global__ __launch_bounds__(256) void post_final(const float* __restrict__ acc,
                                                  const float* __restrict__ b, // [R,OUT]
                                                  float* __restrict__ out) {
    int i = blockIdx.x * blockDim.x + threadIdx.x;
    if (i >= N_NODES * OUT_DIM) return;
    int c = i & (OUT_DIM - 1);
    out[i] = (acc[i] + b[c] + b[OUT_DIM + c] + b[2 * OUT_DIM + c]) * (1.0f / 3.0f);
}

// ================================================================ host
static inline int cdiv(int a, int b) { return (a + b - 1) / b; }

extern "C" void kernel_launch(void* const* d_in, const int* in_sizes, int n_in,
                              void* d_out, int out_size, void* d_ws, size_t ws_size,
                              hipStream_t stream) {
    (void)in_sizes; (void)n_in; (void)out_size; (void)ws_size;

    const float* x       = (const float*)d_in[0];
    const int*   ei[3]   = { (const int*)d_in[1], (const int*)d_in[2], (const int*)d_in[3] };
    const float* W[2]    = { (const float*)d_in[4], (const float*)d_in[8]  };
    const float* asrc[2] = { (const float*)d_in[5], (const float*)d_in[9]  };
    const float* adst[2] = { (const float*)d_in[6], (const float*)d_in[10] };
    const float* bias[2] = { (const float*)d_in[7], (const float*)d_in[11] };
    float* out = (float*)d_out;

    // workspace carve-up (floats)
    float* ws      = (float*)d_ws;
    float* acc     = ws;                               // N*64
    float* h_buf   = acc    + (size_t)N_NODES * OUT_DIM; // N*64
    float* h1      = h_buf  + (size_t)N_NODES * OUT_DIM; // N*64
    float* al_src  = h1     + (size_t)N_NODES * OUT_DIM; // N
    float* al_dst  = al_src + N_NODES;                   // N
    float* mbuf    = al_dst + N_NODES;                   // N
    float* den     = mbuf   + N_NODES;                   // N
    float* e_buf   = den    + N_NODES;                   // E
    float* num_buf = e_buf  + N_EDGES;                   // E

    const int BT = 256;
    const int gemm_blocks = cdiv(N_NODES / 16, BT / 32);
    const int nodeW_blocks = cdiv(N_NODES, BT / 32);
    const int edgeT_blocks = cdiv(N_EDGES, BT);
    const int edgeW_blocks = cdiv(N_EDGES, BT / 32);
    const int nf_blocks = cdiv(N_NODES * OUT_DIM, BT);
    const int n_blocks = cdiv(N_NODES, BT);

    for (int layer = 0; layer < 2; ++layer) {
        const float* X = (layer == 0) ? x : h1;
        const int Wstride = (layer == 0) ? IN_DIM * OUT_DIM : OUT_DIM * OUT_DIM;

        fill_f32<<<nf_blocks, BT, 0, stream>>>(acc, 0.0f, N_NODES * OUT_DIM);

        for (int r = 0; r < R_REL; ++r) {
            // h_r = X @ W[layer][r]
            if (layer == 0)
                gemm_wmma_f32<IN_DIM><<<gemm_blocks, BT, 0, stream>>>(
                    X, W[0] + (size_t)r * Wstride, h_buf);
            else
                gemm_wmma_f32<OUT_DIM><<<gemm_blocks, BT, 0, stream>>>(
                    X, W[1] + (size_t)r * Wstride, h_buf);

            node_attn_logits<<<nodeW_blocks, BT, 0, stream>>>(
                h_buf, asrc[layer] + r * OUT_DIM, adst[layer] + r * OUT_DIM, al_src, al_dst);

            fill_f32<<<n_blocks, BT, 0, stream>>>(mbuf, -INFINITY, N_NODES);
            fill_f32<<<n_blocks, BT, 0, stream>>>(den, 0.0f, N_NODES);

            const int* src = ei[r];
            const int* dst = ei[r] + N_EDGES;
            edge_logits_max<<<edgeT_blocks, BT, 0, stream>>>(src, dst, al_src, al_dst, e_buf, mbuf);
            edge_exp_sum<<<edgeT_blocks, BT, 0, stream>>>(dst, e_buf, mbuf, num_buf, den);
            edge_scatter<<<edgeW_blocks, BT, 0, stream>>>(src, dst, num_buf, den, h_buf, acc);
        }

        if (layer == 0)
            post_norm_relu<<<nodeW_blocks, BT, 0, stream>>>(acc, bias[0], h1);
        else
            post_final<<<nf_blocks, BT, 0, stream>>>(acc, bias[1], out);
    }
}